// ContrastiveComplexGNN_9964324127199
// MI455X (gfx1250) — compile-verified
//
#include <hip/hip_runtime.h>
#include <hip/hip_bf16.h>

#define Bdim 4
#define Ndim 8192
#define NODE 5
#define HID 128
#define PROJ 256
#define KTOT 7
#define BN (Bdim * Ndim)
#define NEDGE (BN * KTOT)
#define FINF __builtin_inff()

typedef __attribute__((ext_vector_type(2))) float v2f;
typedef __attribute__((ext_vector_type(8))) float v8f;

// D = A(16x4, f32) * B(4x16, f32) + C(16x16, f32) on the CDNA5 matrix pipe.
__device__ __forceinline__ v8f wmma4(v2f a, v2f b, v8f c) {
  return __builtin_amdgcn_wmma_f32_16x16x4_f32(
      /*neg_a=*/false, a, /*neg_b=*/false, b,
      /*c_mod=*/(short)0, c, /*reuse_a=*/false, /*reuse_b=*/false);
}

// ---------------- q normalization: qn = q / max(||q||, 1e-8) ----------------
__global__ void qn_kernel(const float* __restrict__ x, float* __restrict__ qn) {
  int n = blockIdx.x * 256 + threadIdx.x;
  if (n >= BN) return;
  float q0 = x[n * NODE + 3], q1 = x[n * NODE + 4];
  float nr = fmaxf(sqrtf(q0 * q0 + q1 * q1), 1e-8f);
  qn[2 * n] = q0 / nr;
  qn[2 * n + 1] = q1 / nr;
}

// ---------------- kNN edge build: one wave per row, LDS column tiles --------
__global__ __launch_bounds__(256) void topk_kernel(const float* __restrict__ pos,
                                                   const float* __restrict__ qn,
                                                   int* __restrict__ dst) {
  const int TS = 512;
  __shared__ float sp0[TS], sp1[TS], sp2[TS], sq0[TS], sq1[TS];
  int b = blockIdx.y;
  int wave = threadIdx.x >> 5, lane = threadIdx.x & 31;
  int i = blockIdx.x * 8 + wave;
  const float* posb = pos + (size_t)b * Ndim * 3;
  const float* qnb = qn + (size_t)b * Ndim * 2;
  float pi0 = posb[3 * i], pi1 = posb[3 * i + 1], pi2 = posb[3 * i + 2];
  float qi0 = qnb[2 * i], qi1 = qnb[2 * i + 1];

  float s[KTOT];
  int id[KTOT];
#pragma unroll
  for (int k = 0; k < KTOT; k++) { s[k] = FINF; id[k] = 0x7FFFFFFF; }

  for (int t = 0; t < Ndim; t += TS) {
    __syncthreads();
    for (int j = threadIdx.x; j < TS; j += 256) {
      int c = t + j;
      sp0[j] = posb[3 * c]; sp1[j] = posb[3 * c + 1]; sp2[j] = posb[3 * c + 2];
      sq0[j] = qnb[2 * c];  sq1[j] = qnb[2 * c + 1];
    }
    __syncthreads();
    for (int j = lane; j < TS; j += 32) {
      int c = t + j;
      float dx = fabsf(pi0 - sp0[j]); if (dx > 0.5f) dx = 1.0f - dx;
      float dy = fabsf(pi1 - sp1[j]); if (dy > 0.5f) dy = 1.0f - dy;
      float dz = fabsf(pi2 - sp2[j]); if (dz > 0.5f) dz = 1.0f - dz;
      float dist = sqrtf(dx * dx + dy * dy + dz * dz + 1e-12f);
      float sc = dist - 0.5f * (1.0f - (qi0 * sq0[j] + qi1 * sq1[j]));
      if (c == i) sc = FINF;
      if (sc < s[KTOT - 1]) {
        s[KTOT - 1] = sc; id[KTOT - 1] = c;
#pragma unroll
        for (int k = KTOT - 1; k > 0; k--) {
          bool sw = (s[k] < s[k - 1]) || (s[k] == s[k - 1] && id[k] < id[k - 1]);
          if (sw) {
            float ts = s[k]; s[k] = s[k - 1]; s[k - 1] = ts;
            int ti = id[k]; id[k] = id[k - 1]; id[k - 1] = ti;
          }
        }
      }
    }
  }
  // 7-round cross-lane merge; lexicographic (score, index) matches jax top_k ties
  for (int r = 0; r < KTOT; r++) {
    float bs = s[0]; int bi = id[0];
#pragma unroll
    for (int off = 16; off > 0; off >>= 1) {
      float os = __shfl_xor(bs, off, 32);
      int oi = __shfl_xor(bi, off, 32);
      if (os < bs || (os == bs && oi < bi)) { bs = os; bi = oi; }
    }
    if (lane == 0) dst[(size_t)(b * Ndim + i) * KTOT + r] = b * Ndim + bi;
    if (s[0] == bs && id[0] == bi) {  // winner lane pops its head
#pragma unroll
      for (int k = 0; k < KTOT - 1; k++) { s[k] = s[k + 1]; id[k] = id[k + 1]; }
      s[KTOT - 1] = FINF; id[KTOT - 1] = 0x7FFFFFFF;
    }
  }
}

// ---------------- degree / inverse degree ----------------------------------
__global__ void deg_count(const int* __restrict__ dst, float* __restrict__ deg) {
  int e = blockIdx.x * 256 + threadIdx.x;
  if (e < NEDGE) atomicAdd(&deg[dst[e]], 1.0f);
}
__global__ void invdeg_kernel(const float* __restrict__ deg, float* __restrict__ inv) {
  int n = blockIdx.x * 256 + threadIdx.x;
  if (n < BN) inv[n] = 1.0f / fmaxf(deg[n], 1.0f);
}

// ---------------- encoder layer 0 (K=5, plain VALU) -------------------------
__global__ void enc0_kernel(const float* __restrict__ x, const float* __restrict__ w,
                            const float* __restrict__ b, float* __restrict__ out) {
  int idx = blockIdx.x * 256 + threadIdx.x;
  int n = idx >> 7, c = idx & 127;
  float acc = b[c];
#pragma unroll
  for (int d = 0; d < NODE; d++) acc += x[n * NODE + d] * w[d * HID + c];
  out[idx] = fmaxf(acc, 0.0f);
}

// -------- weight repack: Wp[k/2][c] = {W[k][c], W[k+1][c]} (b64 B-frags) ----
__global__ void pack_kernel(const float* __restrict__ W, v2f* __restrict__ Wp,
                            int K2N, int N) {
  int idx = blockIdx.x * 256 + threadIdx.x;
  if (idx >= K2N) return;
  int k2 = idx / N, c = idx - k2 * N;
  v2f v; v.x = W[(2 * k2) * N + c]; v.y = W[(2 * k2 + 1) * N + c];
  Wp[idx] = v;
}

// B fragment for k-step kk, half lh, column c: rows kk+2lh, kk+2lh+1
__device__ __forceinline__ v2f bfrag(const v2f* __restrict__ Wp, int kk, int lh,
                                     int c, int NCOL) {
  return Wp[(size_t)((kk >> 1) + lh) * NCOL + c];
}

// ---------------- fp32 WMMA GEMM: wave computes 32 rows x 64 cols -----------
template <int KDIM, int NCOL, bool RELU>
__global__ __launch_bounds__(256) void gemm_wmma(const float* __restrict__ A,
                                                 const v2f* __restrict__ Wp,
                                                 const float* __restrict__ bias,
                                                 float* __restrict__ out) {
  int wid = blockIdx.x * 8 + (threadIdx.x >> 5);
  int lane = threadIdx.x & 31, lh = lane >> 4, ll = lane & 15;
  const int CS = NCOL / 64;
  int r0 = (wid / CS) * 32;
  int c0 = (wid % CS) * 64;
  const float* Arow0 = A + (size_t)(r0 + ll) * KDIM + lh * 2;
  const float* Arow1 = Arow0 + (size_t)16 * KDIM;
  v8f acc0[4] = {}, acc1[4] = {};
  for (int kk = 0; kk < KDIM; kk += 4) {
    v2f a0 = *(const v2f*)(Arow0 + kk);
    v2f a1 = *(const v2f*)(Arow1 + kk);
#pragma unroll
    for (int j = 0; j < 4; j++) {
      v2f bf = bfrag(Wp, kk, lh, c0 + j * 16 + ll, NCOL);
      acc0[j] = wmma4(a0, bf, acc0[j]);
      acc1[j] = wmma4(a1, bf, acc1[j]);
    }
  }
#pragma unroll
  for (int j = 0; j < 4; j++) {
    int c = c0 + j * 16 + ll;
    float bv = bias[c];
#pragma unroll
    for (int v = 0; v < 8; v++) {
      int r = r0 + v + lh * 8;
      float v0 = acc0[j][v] + bv;
      float v1 = acc1[j][v] + bv;
      if (RELU) { v0 = fmaxf(v0, 0.0f); v1 = fmaxf(v1, 0.0f); }
      out[(size_t)r * NCOL + c] = v0;
      out[(size_t)(r + 16) * NCOL + c] = v1;
    }
  }
}

// ---------------- message GEMM: gather [h[src],h[dst]], scatter-add to agg --
__global__ __launch_bounds__(256) void msg_wmma(const float* __restrict__ h,
                                                const v2f* __restrict__ Wp,
                                                const float* __restrict__ bias,
                                                const int* __restrict__ dst,
                                                float* __restrict__ agg) {
  int wid = blockIdx.x * 8 + (threadIdx.x >> 5);
  int lane = threadIdx.x & 31, lh = lane >> 4, ll = lane & 15;
  int r0 = (wid >> 1) * 32;
  int c0 = (wid & 1) * 64;
  int e0 = r0 + ll, e1 = e0 + 16;
  const float* hs0 = h + (size_t)(e0 / KTOT) * HID + lh * 2;  // src = e / 7
  const float* hs1 = h + (size_t)(e1 / KTOT) * HID + lh * 2;
  const float* hd0 = h + (size_t)dst[e0] * HID + lh * 2;
  const float* hd1 = h + (size_t)dst[e1] * HID + lh * 2;
  v8f acc0[4] = {}, acc1[4] = {};
  for (int kk = 0; kk < 2 * HID; kk += 4) {
    v2f a0, a1;
    if (kk < HID) { a0 = *(const v2f*)(hs0 + kk); a1 = *(const v2f*)(hs1 + kk); }
    else          { a0 = *(const v2f*)(hd0 + kk - HID); a1 = *(const v2f*)(hd1 + kk - HID); }
#pragma unroll
    for (int j = 0; j < 4; j++) {
      v2f bf = bfrag(Wp, kk, lh, c0 + j * 16 + ll, HID);
      acc0[j] = wmma4(a0, bf, acc0[j]);
      acc1[j] = wmma4(a1, bf, acc1[j]);
    }
  }
#pragma unroll
  for (int v = 0; v < 8; v++) {
    int r = r0 + v + lh * 8;
    int d0 = dst[r], d1 = dst[r + 16];
#pragma unroll
    for (int j = 0; j < 4; j++) {
      int c = c0 + j * 16 + ll;
      float bv = bias[c];
      atomicAdd(&agg[(size_t)d0 * HID + c], fmaxf(acc0[j][v] + bv, 0.0f));
      atomicAdd(&agg[(size_t)d1 * HID + c], fmaxf(acc1[j][v] + bv, 0.0f));
    }
  }
}

// ---------------- update GEMM: A=[h, agg/deg], h_out = h_in + relu(...) -----
__global__ __launch_bounds__(256) void upd_wmma(const float* __restrict__ hin,
                                                const float* __restrict__ agg,
                                                const float* __restrict__ invd,
                                                const v2f* __restrict__ Wp,
                                                const float* __restrict__ bias,
                                                float* __restrict__ hout) {
  int wid = blockIdx.x * 8 + (threadIdx.x >> 5);
  int lane = threadIdx.x & 31, lh = lane >> 4, ll = lane & 15;
  int r0 = (wid >> 1) * 32;
  int c0 = (wid & 1) * 64;
  int n0 = r0 + ll, n1 = n0 + 16;
  const float* hr0 = hin + (size_t)n0 * HID + lh * 2;
  const float* hr1 = hin + (size_t)n1 * HID + lh * 2;
  const float* ar0 = agg + (size_t)n0 * HID + lh * 2;
  const float* ar1 = agg + (size_t)n1 * HID + lh * 2;
  float idg0 = invd[n0], idg1 = invd[n1];
  v8f acc0[4] = {}, acc1[4] = {};
  for (int kk = 0; kk < 2 * HID; kk += 4) {
    v2f a0, a1;
    if (kk < HID) { a0 = *(const v2f*)(hr0 + kk); a1 = *(const v2f*)(hr1 + kk); }
    else {
      a0 = *(const v2f*)(ar0 + kk - HID); a0.x *= idg0; a0.y *= idg0;
      a1 = *(const v2f*)(ar1 + kk - HID); a1.x *= idg1; a1.y *= idg1;
    }
#pragma unroll
    for (int j = 0; j < 4; j++) {
      v2f bf = bfrag(Wp, kk, lh, c0 + j * 16 + ll, HID);
      acc0[j] = wmma4(a0, bf, acc0[j]);
      acc1[j] = wmma4(a1, bf, acc1[j]);
    }
  }
#pragma unroll
  for (int j = 0; j < 4; j++) {
    int c = c0 + j * 16 + ll;
    float bv = bias[c];
#pragma unroll
    for (int v = 0; v < 8; v++) {
      int r = r0 + v + lh * 8;
      hout[(size_t)r * HID + c] =
          hin[(size_t)r * HID + c] + fmaxf(acc0[j][v] + bv, 0.0f);
      hout[(size_t)(r + 16) * HID + c] =
          hin[(size_t)(r + 16) * HID + c] + fmaxf(acc1[j][v] + bv, 0.0f);
    }
  }
}

extern "C" void kernel_launch(void* const* d_in, const int* in_sizes, int n_in,
                              void* d_out, int out_size, void* d_ws, size_t ws_size,
                              hipStream_t stream) {
  const float* x = (const float*)d_in[0];
  const float* pos = (const float*)d_in[1];
  const float* enc_w0 = (const float*)d_in[2];
  const float* enc_b0 = (const float*)d_in[3];
  const float* enc_w1 = (const float*)d_in[4];
  const float* enc_b1 = (const float*)d_in[5];
  const float* msg_w = (const float*)d_in[6];
  const float* msg_b = (const float*)d_in[7];
  const float* upd_w = (const float*)d_in[8];
  const float* upd_b = (const float*)d_in[9];
  const float* proj_w0 = (const float*)d_in[10];
  const float* proj_b0 = (const float*)d_in[11];
  const float* proj_w1 = (const float*)d_in[12];
  const float* proj_b1 = (const float*)d_in[13];

  float* ws = (float*)d_ws;
  size_t o = 0;
  float* qn = ws + o;   o += (size_t)BN * 2;
  float* deg = ws + o;  o += BN;
  float* invd = ws + o; o += BN;
  float* tmp = ws + o;  o += (size_t)BN * HID;
  float* hA = ws + o;   o += (size_t)BN * HID;
  float* hB = ws + o;   o += (size_t)BN * HID;
  float* agg = ws + o;  o += (size_t)BN * HID;
  // packed weights (v2f = adjacent-k pair per column)
  v2f* wp_enc1 = (v2f*)(ws + o);  o += (size_t)HID * HID;       // 128x128
  v2f* wp_msg  = (v2f*)(ws + o);  o += (size_t)3 * 2 * HID * HID;  // 3 x 256x128
  v2f* wp_upd  = (v2f*)(ws + o);  o += (size_t)3 * 2 * HID * HID;
  v2f* wp_p0   = (v2f*)(ws + o);  o += (size_t)HID * HID;
  v2f* wp_p1   = (v2f*)(ws + o);  o += (size_t)HID * PROJ;      // 128x256
  int* dst = (int*)(ws + o);

  float* zout = (float*)d_out;
  float* hfin = zout + (size_t)BN * PROJ;

  // weight repacks (one-time, tiny)
  pack_kernel<<<(HID / 2 * HID + 255) / 256, 256, 0, stream>>>(enc_w1, wp_enc1, HID / 2 * HID, HID);
  pack_kernel<<<(3 * HID * HID + 255) / 256, 256, 0, stream>>>(msg_w, wp_msg, 3 * HID * HID, HID);
  pack_kernel<<<(3 * HID * HID + 255) / 256, 256, 0, stream>>>(upd_w, wp_upd, 3 * HID * HID, HID);
  pack_kernel<<<(HID / 2 * HID + 255) / 256, 256, 0, stream>>>(proj_w0, wp_p0, HID / 2 * HID, HID);
  pack_kernel<<<(HID / 2 * PROJ + 255) / 256, 256, 0, stream>>>(proj_w1, wp_p1, HID / 2 * PROJ, PROJ);

  qn_kernel<<<BN / 256, 256, 0, stream>>>(x, qn);
  topk_kernel<<<dim3(Ndim / 8, Bdim), 256, 0, stream>>>(pos, qn, dst);
  hipMemsetAsync(deg, 0, (size_t)BN * 4, stream);
  deg_count<<<NEDGE / 256, 256, 0, stream>>>(dst, deg);
  invdeg_kernel<<<BN / 256, 256, 0, stream>>>(deg, invd);

  enc0_kernel<<<(BN * HID) / 256, 256, 0, stream>>>(x, enc_w0, enc_b0, tmp);
  gemm_wmma<HID, HID, true><<<(BN / 32 * 2) / 8, 256, 0, stream>>>(tmp, wp_enc1, enc_b1, hA);

  const float* hin = hA;
  float* ho = hB;
  for (int l = 0; l < 3; l++) {
    hipMemsetAsync(agg, 0, (size_t)BN * HID * 4, stream);
    msg_wmma<<<(NEDGE / 32 * 2) / 8, 256, 0, stream>>>(
        hin, wp_msg + (size_t)l * HID * HID, msg_b + (size_t)l * HID, dst, agg);
    upd_wmma<<<(BN / 32 * 2) / 8, 256, 0, stream>>>(
        hin, agg, invd, wp_upd + (size_t)l * HID * HID, upd_b + (size_t)l * HID, ho);
    const float* t = hin; hin = ho; ho = (float*)t;
  }

  gemm_wmma<HID, HID, true><<<(BN / 32 * 2) / 8, 256, 0, stream>>>(hin, wp_p0, proj_b0, tmp);
  gemm_wmma<HID, PROJ, false><<<(BN / 32 * 4) / 8, 256, 0, stream>>>(tmp, wp_p1, proj_b1, zout);
  hipMemcpyAsync(hfin, hin, (size_t)BN * HID * 4, hipMemcpyDeviceToDevice, stream);
}